// DiscoveryMemorywithDynamicThreshold_88596585382831
// MI455X (gfx1250) — compile-verified
//
#include <hip/hip_runtime.h>
#include <hip/hip_bf16.h>

typedef __attribute__((ext_vector_type(2))) float v2f;
typedef __attribute__((ext_vector_type(8))) float v8f;

#define MEM_SLOTS    100
#define PADDED_SLOTS 112   // multiple of 16 for WMMA tiling; rows >= ptr are zero
#define C_IN         512
#define C_OUT        256
#define HW           4096
#define BATCH        32

// ---------------------------------------------------------------------------
// Kernel 1: proj[b,m,hw] = sum_c W[m,c]*feats[b,c,hw] + bias[m]
// One wave per 16-pixel column; wave owns full M=256 (16 WMMA C-tiles) so
// feats is read exactly once from HBM. f32 WMMA 16x16x4.
// A layout (ISA 7.12.2): lane(0-15)=M row, vgpr j -> K=j, upper half K=j+2.
// B layout (symmetric):  lane(0-15)=N col, vgpr j -> K=j, upper half K=j+2.
// C layout: vgpr r, lane l -> M = r + 8*(l/16), N = l%16.
// ---------------------------------------------------------------------------
__global__ __launch_bounds__(32) void k_proj(const float* __restrict__ feats,
                                             const float* __restrict__ W,
                                             const float* __restrict__ bias,
                                             float* __restrict__ out)
{
    const int lane = threadIdx.x;
    const int lo = lane & 15, hi = lane >> 4;
    const int n0  = blockIdx.x << 4;          // global pixel index (b*4096 + hw)
    const int b   = n0 >> 12;
    const int hw0 = n0 & 4095;

    const v8f vzero = {0.f,0.f,0.f,0.f,0.f,0.f,0.f,0.f};
    v8f acc[16];
#pragma unroll
    for (int t = 0; t < 16; ++t) acc[t] = vzero;

    const float* fbase = feats + ((size_t)b * C_IN << 12) + hw0 + lo;

    for (int k = 0; k < C_IN; k += 4) {
        const int c0 = k + 2 * hi;
        v2f bv;
        bv.x = fbase[(size_t)c0 << 12];
        bv.y = fbase[(size_t)(c0 + 1) << 12];
#pragma unroll
        for (int t = 0; t < 16; ++t) {
            const int m = t * 16 + lo;
            v2f av;
            av.x = W[m * C_IN + c0];
            av.y = W[m * C_IN + c0 + 1];
            acc[t] = __builtin_amdgcn_wmma_f32_16x16x4_f32(
                false, av, false, bv, (short)0, acc[t], false, false);
        }
    }

#pragma unroll
    for (int t = 0; t < 16; ++t) {
#pragma unroll
        for (int r = 0; r < 8; ++r) {
            const int m = t * 16 + r + 8 * hi;
            out[(((size_t)b * 512 + m) << 12) + hw0 + lo] = acc[t][r] + bias[m];
        }
    }
}

// ---------------------------------------------------------------------------
// Kernel 2: pooled[b,c] = mean_hw( proj[b,c,hw] * preds[b,hw] )
// One 256-thread block per (b,c).
// ---------------------------------------------------------------------------
__global__ __launch_bounds__(256) void k_pool(const float* __restrict__ out,   // proj lives in channels 0..255
                                              const float* __restrict__ preds,
                                              float* __restrict__ pooled)
{
    __shared__ float red[256];
    const int b = blockIdx.x >> 8;
    const int c = blockIdx.x & 255;
    const int t = threadIdx.x;
    const float* p = out + (((size_t)b * 512 + c) << 12);
    const float* q = preds + ((size_t)b << 12);
    float s = 0.f;
    for (int i = t; i < HW; i += 256) s += p[i] * q[i];
    red[t] = s; __syncthreads();
    for (int w = 128; w > 0; w >>= 1) { if (t < w) red[t] += red[t + w]; __syncthreads(); }
    if (t == 0) pooled[b * C_OUT + c] = red[0] * (1.0f / HW);
}

// ---------------------------------------------------------------------------
// Kernel 3: sequential codebook scan (single block, 256 threads = 8 waves).
// memw is 112x256, rows >= ptr stay zero (WMMA padding).
// ---------------------------------------------------------------------------
__global__ __launch_bounds__(256) void k_mem(const float* __restrict__ pooled,
                                             const int* __restrict__ epoch_p,
                                             float* __restrict__ memw,
                                             int* __restrict__ ptr_out)
{
    __shared__ float xs[256];
    __shared__ float red[256];
    __shared__ float sims[128];
    __shared__ int s_ptr, s_idx, s_doema;
    const int t = threadIdx.x;
    const int lane = t & 31, w = t >> 5;
    const float thr = ((float)(*epoch_p) * 0.1f - 2.0f) * (0.4f / 13.0f) + 0.3f;

    for (int i = t; i < PADDED_SLOTS * C_OUT; i += 256) memw[i] = 0.f;
    if (t == 0) s_ptr = 0;
    __syncthreads();

    for (int b = 0; b < BATCH; ++b) {
        const float x = pooled[b * C_OUT + t];
        xs[t] = x; red[t] = x * x;
        __syncthreads();
        for (int s = 128; s > 0; s >>= 1) { if (t < s) red[t] += red[t + s]; __syncthreads(); }
        const float nx = sqrtf(red[0]);
        const int cptr = s_ptr;

        // cosine sims: each wave handles slots p = w, w+8, ... ; lane-strided dot
        for (int p = w; p < cptr; p += 8) {
            float d = 0.f, ss = 0.f;
            for (int c = lane; c < C_OUT; c += 32) {
                const float m = memw[p * C_OUT + c];
                d += m * xs[c]; ss += m * m;
            }
#pragma unroll
            for (int o = 16; o > 0; o >>= 1) {
                d  += __shfl_xor(d,  o, 32);
                ss += __shfl_xor(ss, o, 32);
            }
            if (lane == 0) {
                const float dn = (ss == 0.f) ? 1.f : sqrtf(ss);
                sims[p] = d / (dn * nx);
            }
        }
        __syncthreads();
        if (t == 0) {
            int best = 0; float bv = -__builtin_inff();
            for (int p = 0; p < cptr; ++p)
                if (sims[p] > bv) { bv = sims[p]; best = p; }
            s_idx = best;
            s_doema = (cptr > 0) && (bv >= thr);
        }
        __syncthreads();
        if (s_doema) {
            memw[s_idx * C_OUT + t] = memw[s_idx * C_OUT + t] * 0.9f + 0.1f * x;
        } else {
            memw[s_ptr * C_OUT + t] = x;
        }
        __syncthreads();
        if (t == 0 && !s_doema) s_ptr++;
        __syncthreads();
    }
    if (t == 0) *ptr_out = s_ptr;
}

// ---------------------------------------------------------------------------
// Kernel 4: memory cross-attention. One wave per 16-pixel column.
//   logits[112,16] = memw[112,256] @ proj[256,16]   (WMMA, 7 M-tiles, K=256)
//   softmax over valid slots (per pixel: lanes l and l+16 combined via shfl)
//   aug[256,16]   = memw^T[256,112] @ attn[112,16]  (WMMA, 16 M-tiles, K=112)
// attn staged in LDS so B-operand lanes can gather the WMMA layout directly.
// ---------------------------------------------------------------------------
__global__ __launch_bounds__(32) void k_attn(const float* __restrict__ memw,
                                             const int* __restrict__ ptr_p,
                                             float* __restrict__ out)
{
    __shared__ float attn_s[PADDED_SLOTS * 16];
    const int lane = threadIdx.x;
    const int lo = lane & 15, hi = lane >> 4;
    const int n0  = blockIdx.x << 4;
    const int b   = n0 >> 12;
    const int hw0 = n0 & 4095;
    const int ptr = *ptr_p;

    const v8f vzero = {0.f,0.f,0.f,0.f,0.f,0.f,0.f,0.f};

    // ---- logits ----
    v8f lacc[7];
#pragma unroll
    for (int t = 0; t < 7; ++t) lacc[t] = vzero;

    const float* pbase = out + ((size_t)b * 512 << 12) + hw0 + lo;  // proj channels 0..255
    for (int k = 0; k < C_OUT; k += 4) {
        const int c0 = k + 2 * hi;
        v2f bv;
        bv.x = pbase[(size_t)c0 << 12];
        bv.y = pbase[(size_t)(c0 + 1) << 12];
#pragma unroll
        for (int t = 0; t < 7; ++t) {
            const int p = t * 16 + lo;
            v2f av;
            av.x = memw[p * C_OUT + c0];
            av.y = memw[p * C_OUT + c0 + 1];
            lacc[t] = __builtin_amdgcn_wmma_f32_16x16x4_f32(
                false, av, false, bv, (short)0, lacc[t], false, false);
        }
    }

    // ---- softmax over slots (pixel n = lo lives in lanes lo and lo+16) ----
    float mx = -__builtin_inff();
#pragma unroll
    for (int t = 0; t < 7; ++t)
#pragma unroll
        for (int r = 0; r < 8; ++r) {
            const int p = t * 16 + r + 8 * hi;
            if (p < ptr) mx = fmaxf(mx, lacc[t][r]);
        }
    mx = fmaxf(mx, __shfl_xor(mx, 16, 32));
    float sm = 0.f;
#pragma unroll
    for (int t = 0; t < 7; ++t)
#pragma unroll
        for (int r = 0; r < 8; ++r) {
            const int p = t * 16 + r + 8 * hi;
            const float e = (p < ptr) ? __expf(lacc[t][r] - mx) : 0.f;
            lacc[t][r] = e; sm += e;
        }
    sm += __shfl_xor(sm, 16, 32);
    const float inv = 1.0f / sm;
#pragma unroll
    for (int t = 0; t < 7; ++t)
#pragma unroll
        for (int r = 0; r < 8; ++r) {
            const int p = t * 16 + r + 8 * hi;
            attn_s[p * 16 + lo] = lacc[t][r] * inv;
        }
    __syncthreads();

    // ---- aug = memw^T @ attn ----
    v8f acc[16];
#pragma unroll
    for (int t = 0; t < 16; ++t) acc[t] = vzero;

    for (int p0 = 0; p0 < PADDED_SLOTS; p0 += 4) {
        const int pk = p0 + 2 * hi;
        v2f bv;
        bv.x = attn_s[pk * 16 + lo];
        bv.y = attn_s[(pk + 1) * 16 + lo];
#pragma unroll
        for (int t = 0; t < 16; ++t) {
            const int c = t * 16 + lo;
            v2f av;
            av.x = memw[pk * C_OUT + c];
            av.y = memw[(pk + 1) * C_OUT + c];
            acc[t] = __builtin_amdgcn_wmma_f32_16x16x4_f32(
                false, av, false, bv, (short)0, acc[t], false, false);
        }
    }

#pragma unroll
    for (int t = 0; t < 16; ++t)
#pragma unroll
        for (int r = 0; r < 8; ++r) {
            const int m = 256 + t * 16 + r + 8 * hi;
            out[(((size_t)b * 512 + m) << 12) + hw0 + lo] = acc[t][r];
        }
}

// ---------------------------------------------------------------------------
extern "C" void kernel_launch(void* const* d_in, const int* in_sizes, int n_in,
                              void* d_out, int out_size, void* d_ws, size_t ws_size,
                              hipStream_t stream)
{
    (void)in_sizes; (void)n_in; (void)out_size; (void)ws_size;
    const float* feats = (const float*)d_in[0];
    const float* preds = (const float*)d_in[1];
    const float* W     = (const float*)d_in[2];
    const float* bias  = (const float*)d_in[3];
    const int*   epoch = (const int*)d_in[4];
    float* out = (float*)d_out;

    float* pooled = (float*)d_ws;                         // 32*256 f32
    float* memw   = pooled + BATCH * C_OUT;               // 112*256 f32 (zero-padded)
    int*   ptrp   = (int*)(memw + PADDED_SLOTS * C_OUT);  // 1 int

    k_proj<<<dim3(BATCH * HW / 16), dim3(32),  0, stream>>>(feats, W, bias, out);
    k_pool<<<dim3(BATCH * C_OUT),   dim3(256), 0, stream>>>(out, preds, pooled);
    k_mem <<<dim3(1),               dim3(256), 0, stream>>>(pooled, epoch, memw, ptrp);
    k_attn<<<dim3(BATCH * HW / 16), dim3(32),  0, stream>>>(memw, ptrp, out);
}